// GAT_Net_72954314490033
// MI455X (gfx1250) — compile-verified
//
#include <hip/hip_runtime.h>
#include <hip/hip_bf16.h>

typedef __attribute__((ext_vector_type(2))) float v2f;
typedef __attribute__((ext_vector_type(8))) float v8f;

#define NEG_SLOPE 0.2f

// ---- monotone ordered encoding for float atomicMax on unsigned ----
__device__ __forceinline__ unsigned enc_ord(float f) {
    unsigned u = __float_as_uint(f);
    return (u & 0x80000000u) ? ~u : (u | 0x80000000u);
}
__device__ __forceinline__ float dec_ord(unsigned k) {
    unsigned u = (k & 0x80000000u) ? (k & 0x7fffffffu) : ~k;
    return __uint_as_float(u);
}

// =====================================================================
// GEMM: C[M x Nn] = A[M x K=256] @ B[256 x Nn], row-major, fp32, via
// V_WMMA_F32_16X16X4_F32. One wave per 16x16 C tile; block = 8 waves
// covering 8 consecutive m-tiles and one n-tile; B K-slice staged in LDS.
// =====================================================================
__global__ __launch_bounds__(256) void gemm_wmma_f32(
    const float* __restrict__ A, const float* __restrict__ B,
    float* __restrict__ C, int Mtiles, int lda, int ldb, int ldc) {
    __shared__ float Bs[256 * 16];            // 16 KB: B[k, n0..n0+15]
    const int n0 = blockIdx.y * 16;
    for (int i = threadIdx.x; i < 256 * 16; i += 256)
        Bs[i] = B[(size_t)(i >> 4) * ldb + n0 + (i & 15)];
    __syncthreads();

    const int mtile = blockIdx.x * 8 + (threadIdx.x >> 5);
    if (mtile >= Mtiles) return;              // wave-uniform
    const int lane = threadIdx.x & 31;
    const int koff = (lane < 16) ? 0 : 2;     // ISA A/B half-wave K split
    const int col  = lane & 15;
    const float* Arow = A + (size_t)(mtile * 16 + col) * lda + koff;
    __builtin_prefetch(Arow + 64, 0, 0);      // global_prefetch_b8, row tail

    v8f acc{};
#pragma unroll 8
    for (int k0 = 0; k0 < 256; k0 += 4) {
        v2f a = *(const v2f*)(Arow + k0);     // A[row, k0+koff .. +1], 8B aligned
        v2f b;
        b.x = Bs[(k0 + koff) * 16 + col];     // B[k0+koff,   n0+col]
        b.y = Bs[(k0 + koff + 1) * 16 + col]; // B[k0+koff+1, n0+col]
        acc = __builtin_amdgcn_wmma_f32_16x16x4_f32(
            false, a, false, b, (short)0, acc, false, false);
    }
    // C/D layout: VGPR r -> row r (lanes 0-15) / row 8+r (lanes 16-31)
    float* Crow = C + (size_t)(mtile * 16) * ldc + n0 + col;
    const int rbase = (lane < 16) ? 0 : 8;
#pragma unroll
    for (int r = 0; r < 8; ++r)
        Crow[(size_t)(rbase + r) * ldc] = acc[r];
}

// =====================================================================
// alpha_s / alpha_d per (node, head): dot(h[n,h,:], a[h,:])
// =====================================================================
__global__ void node_alpha(const float* __restrict__ h,
                           const float* __restrict__ a_s,
                           const float* __restrict__ a_d,
                           float* __restrict__ as_o, float* __restrict__ ad_o,
                           int N, int H, int C) {
    int i = blockIdx.x * blockDim.x + threadIdx.x;
    if (i >= N * H) return;
    int n = i / H, hd = i % H;
    const float* hp = h + (size_t)n * H * C + (size_t)hd * C;
    const float* sp = a_s + (size_t)hd * C;
    const float* dp = a_d + (size_t)hd * C;
    float s0 = 0.f, s1 = 0.f;
    for (int c = 0; c < C; c += 4) {
        float4 hv = *(const float4*)(hp + c);
        float4 av = *(const float4*)(sp + c);
        float4 bv = *(const float4*)(dp + c);
        s0 += hv.x * av.x + hv.y * av.y + hv.z * av.z + hv.w * av.w;
        s1 += hv.x * bv.x + hv.y * bv.y + hv.z * bv.z + hv.w * bv.w;
    }
    as_o[i] = s0;
    ad_o[i] = s1;
}

// =====================================================================
// Edge pass 1: segment max of leaky_relu(as[src]+ad[dst]) over dst
// (emax pre-zeroed; encoded-uint 0 is below any real float's encoding)
// =====================================================================
__global__ void edge_max(const int* __restrict__ ei, int E, int N, int H,
                         const float* __restrict__ as, const float* __restrict__ ad,
                         unsigned* __restrict__ emax) {
    int e = blockIdx.x * blockDim.x + threadIdx.x;
    if (e >= E + N) return;
    int s = (e < E) ? ei[e] : e - E;
    int d = (e < E) ? ei[E + e] : e - E;
    for (int h = 0; h < H; ++h) {
        float v = as[(size_t)s * H + h] + ad[(size_t)d * H + h];
        v = (v > 0.f) ? v : NEG_SLOPE * v;
        atomicMax(&emax[(size_t)d * H + h], enc_ord(v));
    }
}

// Edge pass 2: denom[dst] += exp(e - emax[dst])
__global__ void edge_den(const int* __restrict__ ei, int E, int N, int H,
                         const float* __restrict__ as, const float* __restrict__ ad,
                         const unsigned* __restrict__ emax, float* __restrict__ den) {
    int e = blockIdx.x * blockDim.x + threadIdx.x;
    if (e >= E + N) return;
    int s = (e < E) ? ei[e] : e - E;
    int d = (e < E) ? ei[E + e] : e - E;
    for (int h = 0; h < H; ++h) {
        float v = as[(size_t)s * H + h] + ad[(size_t)d * H + h];
        v = (v > 0.f) ? v : NEG_SLOPE * v;
        atomicAdd(&den[(size_t)d * H + h], __expf(v - dec_ord(emax[(size_t)d * H + h])));
    }
}

// =====================================================================
// Layer-1 aggregation: out[dst, h*64+c] += h1[src, h*64+c] * alpha(e,h)
// One wave per edge; lanes sweep 256 columns coalesced; alpha computed
// once per head on lanes 0..3 and broadcast via shfl.
// =====================================================================
__global__ void edge_agg1(const int* __restrict__ ei, int E, int N,
                          const float* __restrict__ as, const float* __restrict__ ad,
                          const unsigned* __restrict__ emax, const float* __restrict__ den,
                          const float* __restrict__ h1, float* __restrict__ out1) {
    int wave = (blockIdx.x * blockDim.x + threadIdx.x) >> 5;
    int lane = threadIdx.x & 31;
    if (wave >= E + N) return;                // wave-uniform
    int s = (wave < E) ? ei[wave] : wave - E;
    int d = (wave < E) ? ei[E + wave] : wave - E;
    float alpha = 0.f;
    if (lane < 4) {
        float v = as[(size_t)s * 4 + lane] + ad[(size_t)d * 4 + lane];
        v = (v > 0.f) ? v : NEG_SLOPE * v;
        float m = dec_ord(emax[(size_t)d * 4 + lane]);
        alpha = __expf(v - m) / (den[(size_t)d * 4 + lane] + 1e-16f);
    }
    const float* hs = h1 + (size_t)s * 256;
    float* od = out1 + (size_t)d * 256;
#pragma unroll
    for (int j = 0; j < 8; ++j) {
        int c = lane + 32 * j;                // column 0..255, coalesced
        float a = __shfl(alpha, c >> 6, 32);  // head = c/64
        atomicAdd(&od[c], hs[c] * a);
    }
}

// out1 = elu(out1 + b1), in place (layer-1 bias + activation)
__global__ void bias_elu(float* __restrict__ y, const float* __restrict__ b, int total) {
    int i = blockIdx.x * blockDim.x + threadIdx.x;
    if (i >= total) return;
    float v = y[i] + b[i & 255];
    y[i] = (v > 0.f) ? v : (__expf(v) - 1.f);
}

// =====================================================================
// Layer-2 aggregation (H=1, C=16): 16 lanes per edge
// =====================================================================
__global__ void edge_agg2(const int* __restrict__ ei, int E, int N,
                          const float* __restrict__ as, const float* __restrict__ ad,
                          const unsigned* __restrict__ emax, const float* __restrict__ den,
                          const float* __restrict__ h2, float* __restrict__ out) {
    int hw   = (blockIdx.x * blockDim.x + threadIdx.x) >> 4; // half-wave = edge
    int l16  = threadIdx.x & 15;
    if (hw >= E + N) return;
    int s = (hw < E) ? ei[hw] : hw - E;
    int d = (hw < E) ? ei[E + hw] : hw - E;
    float alpha = 0.f;
    if (l16 == 0) {
        float v = as[s] + ad[d];
        v = (v > 0.f) ? v : NEG_SLOPE * v;
        alpha = __expf(v - dec_ord(emax[d])) / (den[d] + 1e-16f);
    }
    alpha = __shfl(alpha, 0, 16);             // broadcast within 16-lane group
    atomicAdd(&out[(size_t)d * 16 + l16], h2[(size_t)s * 16 + l16] * alpha);
}

// out += b2 (mean over H=1 heads is identity)
__global__ void bias_out(float* __restrict__ y, const float* __restrict__ b, int total) {
    int i = blockIdx.x * blockDim.x + threadIdx.x;
    if (i >= total) return;
    y[i] = y[i] + b[i & 15];
}

// =====================================================================
extern "C" void kernel_launch(void* const* d_in, const int* in_sizes, int n_in,
                              void* d_out, int out_size, void* d_ws, size_t ws_size,
                              hipStream_t stream) {
    const float* x   = (const float*)d_in[0];
    const int*   ei  = (const int*)d_in[1];     // [2, E] flat: src then dst
    const float* W1  = (const float*)d_in[3];
    const float* a1s = (const float*)d_in[4];
    const float* a1d = (const float*)d_in[5];
    const float* b1  = (const float*)d_in[6];
    const float* W2  = (const float*)d_in[7];
    const float* a2s = (const float*)d_in[8];
    const float* a2d = (const float*)d_in[9];
    const float* b2  = (const float*)d_in[10];
    const float* Q   = (const float*)d_in[11];

    const int N  = in_sizes[0] / 256;           // 50000
    const int E  = in_sizes[1] / 2;             // 800000
    const int E2 = E + N;                       // + self loops
    const int Mtiles = N / 16;                  // 3125 (exact)

    // ---- workspace carve-up ----
    char* ws = (char*)d_ws;
    float*    h1    = (float*)ws;                  ws += (size_t)N * 256 * 4;
    float*    out1  = (float*)ws;                  ws += (size_t)N * 256 * 4;
    float*    as1   = (float*)ws;                  ws += (size_t)N * 4 * 4;
    float*    ad1   = (float*)ws;                  ws += (size_t)N * 4 * 4;
    unsigned* emax1 = (unsigned*)ws;               ws += (size_t)N * 4 * 4;
    float*    den1  = (float*)ws;                  ws += (size_t)N * 4 * 4;
    float*    h2    = (float*)ws;                  ws += (size_t)N * 16 * 4;
    float*    as2   = (float*)ws;                  ws += (size_t)N * 4;
    float*    ad2   = (float*)ws;                  ws += (size_t)N * 4;
    unsigned* emax2 = (unsigned*)ws;               ws += (size_t)N * 4;
    float*    den2  = (float*)ws;                  ws += (size_t)N * 4;

    float* out = (float*)d_out;                    // [N,16] then Q

    // ---- per-call zero fills (stream-ordered, capture-safe) ----
    hipMemsetAsync(out1,  0, (size_t)N * 256 * 4, stream);
    hipMemsetAsync(emax1, 0, (size_t)N * 4 * 4,   stream);
    hipMemsetAsync(den1,  0, (size_t)N * 4 * 4,   stream);
    hipMemsetAsync(emax2, 0, (size_t)N * 4,       stream);
    hipMemsetAsync(den2,  0, (size_t)N * 4,       stream);
    hipMemsetAsync(out,   0, (size_t)N * 16 * 4,  stream);

    const int mblocks = (Mtiles + 7) / 8;          // 391

    // ---- layer 1 ----
    gemm_wmma_f32<<<dim3(mblocks, 16), 256, 0, stream>>>(x, W1, h1, Mtiles, 256, 256, 256);
    node_alpha<<<(N * 4 + 255) / 256, 256, 0, stream>>>(h1, a1s, a1d, as1, ad1, N, 4, 64);
    edge_max<<<(E2 + 255) / 256, 256, 0, stream>>>(ei, E, N, 4, as1, ad1, emax1);
    edge_den<<<(E2 + 255) / 256, 256, 0, stream>>>(ei, E, N, 4, as1, ad1, emax1, den1);
    edge_agg1<<<(E2 * 32 + 255) / 256, 256, 0, stream>>>(ei, E, N, as1, ad1, emax1, den1, h1, out1);
    bias_elu<<<(N * 256 + 255) / 256, 256, 0, stream>>>(out1, b1, N * 256);

    // ---- layer 2 ----
    gemm_wmma_f32<<<dim3(mblocks, 1), 256, 0, stream>>>(out1, W2, h2, Mtiles, 256, 16, 16);
    node_alpha<<<(N + 255) / 256, 256, 0, stream>>>(h2, a2s, a2d, as2, ad2, N, 1, 16);
    edge_max<<<(E2 + 255) / 256, 256, 0, stream>>>(ei, E, N, 1, as2, ad2, emax2);
    edge_den<<<(E2 + 255) / 256, 256, 0, stream>>>(ei, E, N, 1, as2, ad2, emax2, den2);
    edge_agg2<<<(E2 * 16 + 255) / 256, 256, 0, stream>>>(ei, E, N, as2, ad2, emax2, den2, h2, out);
    bias_out<<<(N * 16 + 255) / 256, 256, 0, stream>>>(out, b2, N * 16);

    // ---- pass-through Q ----
    hipMemcpyAsync(out + (size_t)N * 16, Q, (size_t)in_sizes[11] * 4,
                   hipMemcpyDeviceToDevice, stream);
}